// SemiLoss_60430189855042
// MI455X (gfx1250) — compile-verified
//
#include <hip/hip_runtime.h>

// FocalLoss_semi on MI455X (gfx1250, wave32).
// Shapes: out_seg [48, 11, 36864] f32, seg [48, 36864] i32, label [48] i32.
// Memory-bound gather-reduce (~85 MB @ 23.3 TB/s ~= 3.7 us). WMMA f32 16x16x4
// with a ones B-matrix performs the exact f32 cross-lane reduction on the
// matrix pipe (co-executes with VALU), replacing the shuffle tree.

typedef float v2f __attribute__((ext_vector_type(2)));
typedef float v8f __attribute__((ext_vector_type(8)));

#define NSAMP 48          // B*S = 8*6
#define KCLS  11
#define PPIX  36864       // 192*192
#define TPB   256         // 8 waves of 32
#define PIX_PER_THREAD 8
#define PIX_PER_BLOCK  (TPB * PIX_PER_THREAD)   // 2048
#define CHUNKS (PPIX / PIX_PER_BLOCK)           // 18, exact

__global__ __launch_bounds__(TPB) void focal_partial_kernel(
    const float* __restrict__ logits,   // [NSAMP, KCLS, PPIX]
    const int*   __restrict__ seg,      // [NSAMP, PPIX]
    float*       __restrict__ partials) // [NSAMP * CHUNKS]
{
    const int blk   = blockIdx.x;
    const int n     = blk / CHUNKS;
    const int chunk = blk % CHUNKS;
    const int tid   = threadIdx.x;

    const int*   segp  = seg    + (long long)n * PPIX + chunk * PIX_PER_BLOCK + tid;
    const float* lbase = logits + (long long)n * KCLS * PPIX + chunk * PIX_PER_BLOCK + tid;

    float acc = 0.0f;
#pragma unroll
    for (int i = 0; i < PIX_PER_THREAD; ++i) {
        // streamed once: non-temporal so we don't churn L2
        int   t  = __builtin_nontemporal_load(segp + i * TPB);
        float lp = __builtin_nontemporal_load(lbase + (long long)t * PPIX + i * TPB);
        float pt = __expf(lp);            // pt = exp(logpt) (detached in ref)
        float u  = 1.0f - pt;
        float u2 = u * u;
        float u6 = u2 * u2 * u2;          // gamma = 6
        acc = fmaf(-u6, lp, acc);         // sum of -(1-pt)^6 * logpt
    }

    // ---- wave32 reduction via the f32 matrix pipe ----
    // A (16x4 f32): lanes 0-15 supply K=0 (vgpr0) / K=1 (vgpr1),
    //               lanes 16-31 supply K=2 / K=3. Put acc in vgpr0, 0 in vgpr1.
    // B = ones  =>  D[m][n] = acc[m] + acc[m+16]  (all columns identical).
    v2f a; a[0] = acc;  a[1] = 0.0f;
    v2f b; b[0] = 1.0f; b[1] = 1.0f;
    v8f c = {};
    v8f d = __builtin_amdgcn_wmma_f32_16x16x4_f32(
        /*neg_a=*/false, a, /*neg_b=*/false, b,
        /*c_mod=*/(short)0, c, /*reuse_a=*/false, /*reuse_b=*/false);

    // Per lane: D vgpr r holds row r (lanes 0-15) or row r+8 (lanes 16-31).
    // Summing the 8 regs gives rows 0-7 (lanes<16) or rows 8-15 (lanes>=16).
    float half  = d[0] + d[1] + d[2] + d[3] + d[4] + d[5] + d[6] + d[7];
    float wtot  = half + __shfl_xor(half, 16, 32);   // full 32-lane sum

    __shared__ float s_w[TPB / 32];
    if ((tid & 31) == 0) s_w[tid >> 5] = wtot;
    __syncthreads();
    if (tid == 0) {
        float bs = 0.0f;
#pragma unroll
        for (int w = 0; w < TPB / 32; ++w) bs += s_w[w];
        partials[blk] = bs;
    }
}

__global__ void focal_final_kernel(const float* __restrict__ partials, // [NSAMP*CHUNKS]
                                   const int*   __restrict__ label,    // [NSAMP]
                                   float*       __restrict__ out)      // [3]
{
    __shared__ float s_t[64];
    __shared__ float s_c[64];
    int t = threadIdx.x;
    float ps = 0.0f, m = 0.0f;
    if (t < NSAMP) {
        for (int c = 0; c < CHUNKS; ++c) ps += partials[t * CHUNKS + c];
        m = (label[t] != 0) ? 1.0f : 0.0f;
    }
    s_t[t] = ps * m;
    s_c[t] = m;
    __syncthreads();
    if (t == 0) {
        float tot = 0.0f, cnt = 0.0f;
        for (int i = 0; i < NSAMP; ++i) { tot += s_t[i]; cnt += s_c[i]; }
        float loss = tot / cnt;           // matches reference (NaN/Inf if cnt==0)
        out[0] = loss;                    // loss
        out[1] = 0.0f;                    // recon dummy
        out[2] = loss;                    // loss again
    }
}

extern "C" void kernel_launch(void* const* d_in, const int* in_sizes, int n_in,
                              void* d_out, int out_size, void* d_ws, size_t ws_size,
                              hipStream_t stream) {
    const float* out_seg = (const float*)d_in[0];  // [8,6,11,192,192,1] f32
    const int*   seg     = (const int*)  d_in[1];  // [8,6,192,192]
    const int*   label   = (const int*)  d_in[2];  // [8,6]
    float*       out     = (float*)d_out;
    float*       parts   = (float*)d_ws;           // NSAMP*CHUNKS = 864 floats

    focal_partial_kernel<<<NSAMP * CHUNKS, TPB, 0, stream>>>(out_seg, seg, parts);
    focal_final_kernel<<<1, 64, 0, stream>>>(parts, label, out);
}